// ScoreTangentMetric_63453846831507
// MI455X (gfx1250) — compile-verified
//
#include <hip/hip_runtime.h>

// ScoreTangentMetric, MI455X / gfx1250, wave32.
//
// out[b] = ||xd_b||^2 + (s_b . xd_b)^2,   s = x @ W^T
//
// GEMM via V_WMMA_F32_16X16X4_F32 (f32 A/B/C -> exact reference precision),
// fused per-row reduction epilogue through LDS.

typedef float v2f __attribute__((ext_vector_type(2)));
typedef float v8f __attribute__((ext_vector_type(8)));

#define DDIM 512
#define LDS_STRIDE 516   // 512 + 4 pad to rotate LDS banks between rows

__global__ __launch_bounds__(256) void stm_wmma_kernel(
    const float* __restrict__ x,    // [B, 512]
    const float* __restrict__ xd,   // [B, 512]
    const float* __restrict__ W,    // [512, 512]
    float* __restrict__ out)        // [B]
{
    __shared__ float s_tile[16 * LDS_STRIDE];  // 16 x 512 tile of s
    __shared__ float s_dot[16];
    __shared__ float s_norm[16];

    const int tid  = threadIdx.x;
    const int wave = tid >> 5;      // 0..7
    const int lane = tid & 31;
    const int half = lane >> 4;     // which 16-lane half of the wave
    const int nl   = lane & 15;     // row/col index within the 16-wide tile
    const int m0   = blockIdx.x * 16;

    if (tid < 16) { s_dot[tid] = 0.0f; s_norm[tid] = 0.0f; }
    __syncthreads();

    // ---- GEMM: C[16x512] = x[m0:m0+16, :] * W^T, distributed over 8 waves ----
    // A fragment (16x4 f32): lane nl holds row m0+nl, VGPRs hold K = 2*half, 2*half+1.
    // B fragment (4x16 f32): lane nl holds col n0+nl, VGPRs hold K = 2*half, 2*half+1,
    //                        with B[k][n] = W[n][k]  (x @ W^T contraction).
    const float* xrow = x + (size_t)(m0 + nl) * DDIM;

    v8f acc[4] = {};

    for (int k = 0; k < DDIM; k += 4) {
        v2f a = *(const v2f*)(xrow + k + 2 * half);   // 8B-aligned float2
#pragma unroll
        for (int t = 0; t < 4; ++t) {
            const int n0 = (wave + 8 * t) * 16;       // this wave's N tiles
            v2f b = *(const v2f*)(W + (size_t)(n0 + nl) * DDIM + k + 2 * half);
            // 8 args: (neg_a, A, neg_b, B, c_mod, C, reuse_a, reuse_b)
            acc[t] = __builtin_amdgcn_wmma_f32_16x16x4_f32(
                false, a, false, b, (short)0, acc[t], false, false);
        }
    }

    // ---- spill s tile to LDS (C/D layout: VGPR r -> row r + 8*half, col nl) ----
#pragma unroll
    for (int t = 0; t < 4; ++t) {
        const int n = (wave + 8 * t) * 16 + nl;
#pragma unroll
        for (int r = 0; r < 8; ++r) {
            s_tile[(r + 8 * half) * LDS_STRIDE + n] = acc[t][r];
        }
    }
    __syncthreads();

    // ---- fused reduction: 16 threads per batch row ----
    const int row = tid >> 4;   // 0..15  (batch row m0+row)
    const int cg  = tid & 15;   // column group
    const float* xdrow = xd + (size_t)(m0 + row) * DDIM;

    float pdot = 0.0f, pnorm = 0.0f;
#pragma unroll 4
    for (int i = 0; i < DDIM / 16; ++i) {
        const int c = cg + 16 * i;
        const float xv = xdrow[c];
        pdot  += s_tile[row * LDS_STRIDE + c] * xv;
        pnorm += xv * xv;
    }
    atomicAdd(&s_dot[row],  pdot);   // ds_add_f32
    atomicAdd(&s_norm[row], pnorm);
    __syncthreads();

    if (tid < 16) {
        const float dv = s_dot[tid];
        out[m0 + tid] = s_norm[tid] + dv * dv;
    }
}

extern "C" void kernel_launch(void* const* d_in, const int* in_sizes, int n_in,
                              void* d_out, int out_size, void* d_ws, size_t ws_size,
                              hipStream_t stream) {
    (void)n_in; (void)d_ws; (void)ws_size; (void)out_size;
    const float* x  = (const float*)d_in[0];   // [B, 512]
    const float* xd = (const float*)d_in[1];   // [B, 512]
    const float* W  = (const float*)d_in[2];   // [512, 512]
    float* out = (float*)d_out;                // [B]

    const int B = in_sizes[0] / DDIM;          // 1024
    const int grid = B / 16;                   // one block per 16 batch rows

    stm_wmma_kernel<<<grid, 256, 0, stream>>>(x, xd, W, out);
}